// GIN_747324309861
// MI455X (gfx1250) — compile-verified
//
#include <hip/hip_runtime.h>

#define N_NODES 65536
#define N_EDGES 1048576
#define N_GRAPHS 512
#define HID 64
#define BN_EPS 1e-5f

typedef __attribute__((ext_vector_type(2))) float v2f;
typedef __attribute__((ext_vector_type(8))) float v8f;

// ---------------------------------------------------------------------------
// z[n][c] = h[n][c] for c < C else 0   (z stride fixed at 64)
// ---------------------------------------------------------------------------
__global__ void __launch_bounds__(256)
init_z_kernel(float* __restrict__ z, const float* __restrict__ h, int stride, int C) {
    int idx = blockIdx.x * 256 + threadIdx.x;   // over N_NODES*64
    int n = idx >> 6, c = idx & 63;
    z[idx] = (c < C) ? h[n * stride + c] : 0.0f;
}

// ---------------------------------------------------------------------------
// Edge scatter-add: z[dst] += h[src]   (7-channel and 64-channel variants)
// ---------------------------------------------------------------------------
__global__ void __launch_bounds__(256)
edge_agg7_kernel(float* __restrict__ z, const float* __restrict__ x,
                 const int* __restrict__ src, const int* __restrict__ dst) {
    int e = blockIdx.x * 256 + threadIdx.x;       // exactly N_EDGES threads
    int s = src[e], d = dst[e];
    const float* xs = x + s * 7;
    float* zd = z + (size_t)d * 64;
#pragma unroll
    for (int c = 0; c < 7; ++c) atomicAdd(&zd[c], xs[c]);
}

__global__ void __launch_bounds__(256)
edge_agg64_kernel(float* __restrict__ z, const float* __restrict__ h,
                  const int* __restrict__ src, const int* __restrict__ dst) {
    int idx = blockIdx.x * 256 + threadIdx.x;     // exactly N_EDGES*16 threads
    int e = idx >> 4, q = idx & 15;
    int s = src[e], d = dst[e];
    float4 hv = *(const float4*)(h + (size_t)s * 64 + q * 4);
    float* zd = z + (size_t)d * 64 + q * 4;
    atomicAdd(&zd[0], hv.x);
    atomicAdd(&zd[1], hv.y);
    atomicAdd(&zd[2], hv.z);
    atomicAdd(&zd[3], hv.w);
}

// ---------------------------------------------------------------------------
// WMMA GEMM: Y[N x 64] = A[N x 64(stride)] @ W[Krows x 64] + bias
// Also accumulates per-channel sum / sum-of-squares into stats[128] for BN.
// One wave = 16 output rows x 64 cols (4 WMMA f32 16x16 tiles).
// 256 threads = 8 waves = 128 rows per block; grid = N/128.
// ---------------------------------------------------------------------------
template <int KSTEPS>
__global__ void __launch_bounds__(256)
gin_gemm_wmma(const float* __restrict__ A, const float* __restrict__ W,
              const float* __restrict__ bias, float* __restrict__ Y,
              float* __restrict__ stats, int Krows) {
    __shared__ float sW[64 * 64];
    __shared__ float sSum[64];
    __shared__ float sSq[64];
    int tid = threadIdx.x;

    for (int i = tid; i < 64 * 64; i += 256) {
        int r = i >> 6;
        sW[i] = (r < Krows) ? W[i] : 0.0f;
    }
    if (tid < 64) { sSum[tid] = 0.0f; sSq[tid] = 0.0f; }
    __syncthreads();

    int wave = tid >> 5;
    int lane = tid & 31;
    int half = lane >> 4;     // 0: K pair {0,1}, 1: K pair {2,3} of each 4-step
    int l16  = lane & 15;
    int row0 = blockIdx.x * 128 + wave * 16;

    // A fragment: 16x4 f32 layout — lane l(<16): A[M=l][k0], A[M=l][k0+1];
    // lanes 16-31: A[M=l][k0+2], A[M=l][k0+3]
    const float* arow = A + (size_t)(row0 + l16) * 64 + half * 2;
    v2f afrag[KSTEPS];
#pragma unroll
    for (int kk = 0; kk < KSTEPS; ++kk) {
        afrag[kk].x = arow[kk * 4 + 0];
        afrag[kk].y = arow[kk * 4 + 1];
    }

    v8f acc[4] = {};  // zero accumulators, 4 column tiles (64 output cols)

#pragma unroll
    for (int kk = 0; kk < KSTEPS; ++kk) {
        int krow = kk * 4 + half * 2;
#pragma unroll
        for (int t = 0; t < 4; ++t) {
            // B fragment (4x16): lane n holds col n of B; VGPR0=K row k0, VGPR1=k0+1
            v2f b;
            b.x = sW[(krow + 0) * 64 + t * 16 + l16];
            b.y = sW[(krow + 1) * 64 + t * 16 + l16];
            acc[t] = __builtin_amdgcn_wmma_f32_16x16x4_f32(
                false, afrag[kk], false, b, (short)0, acc[t], false, false);
        }
    }

    // C/D layout: lane = col + 16*(m>=8); VGPR r holds row (half*8 + r)
    int rbase = row0 + half * 8;
#pragma unroll
    for (int t = 0; t < 4; ++t) {
        int col = t * 16 + l16;
        float bv = bias[col];
        float lsum = 0.0f, lsq = 0.0f;
#pragma unroll
        for (int r = 0; r < 8; ++r) {
            float v = acc[t][r] + bv;
            Y[(size_t)(rbase + r) * 64 + col] = v;
            lsum += v;
            lsq += v * v;
        }
        atomicAdd(&sSum[col], lsum);   // LDS float atomic (ds_add_f32)
        atomicAdd(&sSq[col], lsq);
    }
    __syncthreads();
    if (tid < 64) {
        atomicAdd(&stats[tid], sSum[tid]);
        atomicAdd(&stats[64 + tid], sSq[tid]);
    }
}

// ---------------------------------------------------------------------------
// BN finalize: scale/shift per channel from global stats
// ---------------------------------------------------------------------------
__global__ void bn_finalize_kernel(const float* __restrict__ stats,
                                   const float* __restrict__ gamma,
                                   const float* __restrict__ beta,
                                   float* __restrict__ ss) {
    int c = threadIdx.x;   // 64 threads
    const float invN = 1.0f / (float)N_NODES;
    float mean = stats[c] * invN;
    float var  = stats[64 + c] * invN - mean * mean;
    float scale = gamma[c] * rsqrtf(var + BN_EPS);
    ss[c]      = scale;
    ss[64 + c] = beta[c] - mean * scale;
}

// ---------------------------------------------------------------------------
// out = relu(in * scale[c] + shift[c]) elementwise (float4 per thread)
// ---------------------------------------------------------------------------
__global__ void __launch_bounds__(256)
bn_relu_apply_kernel(const float* __restrict__ in, const float* __restrict__ ss,
                     float* __restrict__ out) {
    int idx = blockIdx.x * 256 + threadIdx.x;   // over N_NODES*64/4
    int base = idx * 4;
    int c0 = base & 63;
    float4 v = *(const float4*)(in + base);
    float4 o;
    o.x = fmaxf(fmaf(v.x, ss[c0 + 0], ss[64 + c0 + 0]), 0.0f);
    o.y = fmaxf(fmaf(v.y, ss[c0 + 1], ss[64 + c0 + 1]), 0.0f);
    o.z = fmaxf(fmaf(v.z, ss[c0 + 2], ss[64 + c0 + 2]), 0.0f);
    o.w = fmaxf(fmaf(v.w, ss[c0 + 3], ss[64 + c0 + 3]), 0.0f);
    *(float4*)(out + base) = o;
}

// ---------------------------------------------------------------------------
// global_add_pool: pool[batch[n]][c] += rep[n][c]
// ---------------------------------------------------------------------------
__global__ void __launch_bounds__(256)
pool_agg_kernel(const float* __restrict__ rep, int stride, int C,
                const int* __restrict__ batch, float* __restrict__ pool) {
    int idx = blockIdx.x * 256 + threadIdx.x;   // over N_NODES*64
    int n = idx >> 6, c = idx & 63;
    if (c < C) atomicAdd(&pool[(size_t)batch[n] * 64 + c], rep[(size_t)n * stride + c]);
}

// ---------------------------------------------------------------------------
// Readout MLP on pooled [512 x fi]: relu(W0)+relu(W1)+W2, accumulate score.
// One block per graph, 64 threads.
// ---------------------------------------------------------------------------
__global__ void __launch_bounds__(64)
readout_kernel(const float* __restrict__ pool,
               const float* __restrict__ W0, const float* __restrict__ b0,
               const float* __restrict__ W1, const float* __restrict__ b1,
               const float* __restrict__ W2, const float* __restrict__ b2,
               int fi, float* __restrict__ out) {
    __shared__ float p[64];
    __shared__ float q[64];
    int g = blockIdx.x, t = threadIdx.x;
    p[t] = pool[g * 64 + t];
    __syncthreads();
    float a = b0[t];
    for (int k = 0; k < fi; ++k) a += p[k] * W0[k * 64 + t];
    q[t] = fmaxf(a, 0.0f);
    __syncthreads();
    a = b1[t];
    for (int k = 0; k < 64; ++k) a += q[k] * W1[k * 64 + t];
    __syncthreads();
    p[t] = fmaxf(a, 0.0f);
    __syncthreads();
    if (t < 2) {
        float a2 = b2[t];
        for (int k = 0; k < 64; ++k) a2 += p[k] * W2[k * 2 + t];
        out[g * 2 + t] += a2;   // sequential kernel launches accumulate
    }
}

// ---------------------------------------------------------------------------
extern "C" void kernel_launch(void* const* d_in, const int* in_sizes, int n_in,
                              void* d_out, int out_size, void* d_ws, size_t ws_size,
                              hipStream_t stream) {
    const float* x     = (const float*)d_in[0];
    const int*   src   = (const int*)d_in[1];            // edge_index row 0
    const int*   dstE  = ((const int*)d_in[1]) + N_EDGES; // edge_index row 1
    const int*   batch = (const int*)d_in[2];

    // ---- parameter mapping (auto-detect flatten convention from sizes) ----
    // sorted-key flatten: conv leaves start with bn_b (size 64);
    // insertion-order flatten: conv leaves start with W0 (size 448).
    bool sortedKeys = (in_sizes[3] == 64);
    struct Conv {
        const float *W[3], *b[3], *bn_g[2], *bn_b[2], *obn_g, *obn_b;
    } convs[3];
    for (int l = 0; l < 3; ++l) {
        int s = 3 + 12 * l;
        if (sortedKeys) {
            convs[l].bn_b[0] = (const float*)d_in[s + 0];
            convs[l].bn_b[1] = (const float*)d_in[s + 1];
            convs[l].bn_g[0] = (const float*)d_in[s + 2];
            convs[l].bn_g[1] = (const float*)d_in[s + 3];
            convs[l].W[0] = (const float*)d_in[s + 4];
            convs[l].b[0] = (const float*)d_in[s + 5];
            convs[l].W[1] = (const float*)d_in[s + 6];
            convs[l].b[1] = (const float*)d_in[s + 7];
            convs[l].W[2] = (const float*)d_in[s + 8];
            convs[l].b[2] = (const float*)d_in[s + 9];
            convs[l].obn_b = (const float*)d_in[s + 10];
            convs[l].obn_g = (const float*)d_in[s + 11];
        } else {
            convs[l].W[0] = (const float*)d_in[s + 0];
            convs[l].b[0] = (const float*)d_in[s + 1];
            convs[l].W[1] = (const float*)d_in[s + 2];
            convs[l].b[1] = (const float*)d_in[s + 3];
            convs[l].W[2] = (const float*)d_in[s + 4];
            convs[l].b[2] = (const float*)d_in[s + 5];
            convs[l].bn_g[0] = (const float*)d_in[s + 6];
            convs[l].bn_g[1] = (const float*)d_in[s + 7];
            convs[l].bn_b[0] = (const float*)d_in[s + 8];
            convs[l].bn_b[1] = (const float*)d_in[s + 9];
            convs[l].obn_g = (const float*)d_in[s + 10];
            convs[l].obn_b = (const float*)d_in[s + 11];
        }
    }
    const float *mlpW[4][3], *mlpB[4][3];
    for (int i = 0; i < 4; ++i) {
        int s = 3 + 36 + 6 * i;   // {"W","b"} same order either way
        for (int j = 0; j < 3; ++j) {
            mlpW[i][j] = (const float*)d_in[s + 2 * j];
            mlpB[i][j] = (const float*)d_in[s + 2 * j + 1];
        }
    }

    // ---- workspace layout ----
    const size_t TEN = (size_t)N_NODES * 64;  // 4,194,304 floats
    float* ws    = (float*)d_ws;
    float* zbuf  = ws;                 // ping
    float* t0    = ws + TEN;           // pong
    float* hbuf0 = ws + 2 * TEN;       // hidden_rep[1]
    float* hbuf1 = ws + 3 * TEN;       // hidden_rep[2]
    float* hbuf2 = ws + 4 * TEN;       // hidden_rep[3]
    float* stats = ws + 5 * TEN;       // [128]
    float* ssbuf = stats + 128;        // [128]
    float* pool  = ssbuf + 128;        // [512*64]
    float* hb[3] = {hbuf0, hbuf1, hbuf2};

    const float* hprev = x;
    int stride = 7, C = 7;

    for (int l = 0; l < 3; ++l) {
        // z = h + sum_{j->i} h_j
        init_z_kernel<<<(N_NODES * 64) / 256, 256, 0, stream>>>(zbuf, hprev, stride, C);
        if (C == 7)
            edge_agg7_kernel<<<N_EDGES / 256, 256, 0, stream>>>(zbuf, hprev, src, dstE);
        else
            edge_agg64_kernel<<<(N_EDGES * 16) / 256, 256, 0, stream>>>(zbuf, hprev, src, dstE);

        // lin0 + BN + relu
        hipMemsetAsync(stats, 0, 128 * sizeof(float), stream);
        if (C == 7)
            gin_gemm_wmma<2><<<N_NODES / 128, 256, 0, stream>>>(zbuf, convs[l].W[0], convs[l].b[0], t0, stats, 7);
        else
            gin_gemm_wmma<16><<<N_NODES / 128, 256, 0, stream>>>(zbuf, convs[l].W[0], convs[l].b[0], t0, stats, 64);
        bn_finalize_kernel<<<1, 64, 0, stream>>>(stats, convs[l].bn_g[0], convs[l].bn_b[0], ssbuf);
        bn_relu_apply_kernel<<<(N_NODES * 64) / (256 * 4), 256, 0, stream>>>(t0, ssbuf, t0);

        // lin1 + BN + relu
        hipMemsetAsync(stats, 0, 128 * sizeof(float), stream);
        gin_gemm_wmma<16><<<N_NODES / 128, 256, 0, stream>>>(t0, convs[l].W[1], convs[l].b[1], zbuf, stats, 64);
        bn_finalize_kernel<<<1, 64, 0, stream>>>(stats, convs[l].bn_g[1], convs[l].bn_b[1], ssbuf);
        bn_relu_apply_kernel<<<(N_NODES * 64) / (256 * 4), 256, 0, stream>>>(zbuf, ssbuf, zbuf);

        // lin2 + outer BN + relu -> h_l
        hipMemsetAsync(stats, 0, 128 * sizeof(float), stream);
        gin_gemm_wmma<16><<<N_NODES / 128, 256, 0, stream>>>(zbuf, convs[l].W[2], convs[l].b[2], t0, stats, 64);
        bn_finalize_kernel<<<1, 64, 0, stream>>>(stats, convs[l].obn_g, convs[l].obn_b, ssbuf);
        bn_relu_apply_kernel<<<(N_NODES * 64) / (256 * 4), 256, 0, stream>>>(t0, ssbuf, hb[l]);

        hprev = hb[l];
        stride = 64;
        C = 64;
    }

    // ---- readout over hidden_rep = [x, h1, h2, h3] ----
    hipMemsetAsync(d_out, 0, (size_t)N_GRAPHS * 2 * sizeof(float), stream);
    const float* reps[4]  = {x, hbuf0, hbuf1, hbuf2};
    int strides[4] = {7, 64, 64, 64};
    int Cs[4]      = {7, 64, 64, 64};
    for (int i = 0; i < 4; ++i) {
        hipMemsetAsync(pool, 0, (size_t)N_GRAPHS * 64 * sizeof(float), stream);
        pool_agg_kernel<<<(N_NODES * 64) / 256, 256, 0, stream>>>(reps[i], strides[i], Cs[i], batch, pool);
        readout_kernel<<<N_GRAPHS, 64, 0, stream>>>(pool, mlpW[i][0], mlpB[i][0],
                                                    mlpW[i][1], mlpB[i][1],
                                                    mlpW[i][2], mlpB[i][2],
                                                    Cs[i], (float*)d_out);
    }
}